// PartialCRFLayer_63256278335781
// MI455X (gfx1250) — compile-verified
//
#include <hip/hip_runtime.h>
#include <hip/hip_bf16.h>
#include <math.h>

typedef __attribute__((ext_vector_type(2))) float v2f;
typedef __attribute__((ext_vector_type(8))) float v8f;
typedef __attribute__((ext_vector_type(4))) int   v4i;

#define BATCH 64
#define SEQ 512
#define NIN 1024
#define NTAGS 48
#define NINF_V (-1e8f)

#define KCHUNK 128
#define KPAIRS (KCHUNK / 2)
#define ROWS_PER_BLOCK 64          // 4 waves * 16 rows
#define AS_STRIDE (KCHUNK + 4)     // 132 floats: 8B-aligned rows, conflict-free frag reads
#define WS2_STRIDE 98              // 96 data + 2 pad: 8B-aligned rows, conflict-free stores

#define GLOBAL_AS __attribute__((address_space(1)))
#define LDS_AS    __attribute__((address_space(3)))

#if defined(__has_builtin)
#  if __has_builtin(__builtin_amdgcn_global_load_async_to_lds_b128) && \
      __has_builtin(__builtin_amdgcn_global_load_async_to_lds_b32)
#    define HAVE_ASYNC_LDS 1
#  endif
#endif
#ifndef HAVE_ASYNC_LDS
#  define HAVE_ASYNC_LDS 0
#endif

__device__ __forceinline__ void wait_async_all()
{
#if HAVE_ASYNC_LDS
#  if __has_builtin(__builtin_amdgcn_s_wait_asynccnt)
    __builtin_amdgcn_s_wait_asynccnt(0);
#  else
    asm volatile("s_wait_asynccnt 0" ::: "memory");
#  endif
#endif
}

// ---------------------------------------------------------------------------
// Emissions GEMM: em[m][t] = sum_k x[m][k] * W[t][k] + b[t],  m = b*SEQ + s
// fp32 WMMA 16x16x4, one block = 64 rows x 48 cols, async-LDS staged tiles.
// ---------------------------------------------------------------------------
__global__ __launch_bounds__(128)
void emissions_wmma_kernel(const float* __restrict__ x,
                           const float* __restrict__ W,
                           const float* __restrict__ bias,
                           float* __restrict__ em)
{
    __shared__ float As[ROWS_PER_BLOCK][AS_STRIDE];
    __shared__ float Ws2[KPAIRS][WS2_STRIDE];   // Ws2[kk/2][t*2 + (kk&1)] = W[t][k0+kk]

    const int tid  = threadIdx.x;
    const int lane = tid & 31;
    const int wave = tid >> 5;
    const int m0   = blockIdx.x * ROWS_PER_BLOCK;

    v8f acc0 = {}; v8f acc1 = {}; v8f acc2 = {};

    const int mrow = lane & 15;        // A-fragment row (ISA 16x4 f32 layout)
    const int ko   = (lane >> 4) * 2;  // lanes 0-15 -> K{0,1}; lanes 16-31 -> K{2,3}
    const int ncol = lane & 15;        // B-fragment column

    for (int k0 = 0; k0 < NIN; k0 += KCHUNK) {
        // stage A tile: 64 rows x 128 K, 16B per async op (coalesced along K)
        for (int fi = tid; fi < (ROWS_PER_BLOCK * KCHUNK / 4); fi += 128) {
            const int row = fi >> 5;
            const int c4  = (fi & 31) << 2;
            const float* gp = &x[(size_t)(m0 + row) * NIN + k0 + c4];
            float* sp = &As[row][c4];
#if HAVE_ASYNC_LDS
            __builtin_amdgcn_global_load_async_to_lds_b128(
                (GLOBAL_AS v4i*)gp, (LDS_AS v4i*)sp, 0, 0);
#else
            const float4 v = *reinterpret_cast<const float4*>(gp);
            sp[0] = v.x; sp[1] = v.y; sp[2] = v.z; sp[3] = v.w;
#endif
        }
        // stage W^T chunk, K-pair interleaved so B frags are single b64 loads
        for (int t = 0; t < NTAGS; ++t) {
            const float* gp = &W[(size_t)t * NIN + k0 + tid];
            float* sp = &Ws2[tid >> 1][t * 2 + (tid & 1)];
#if HAVE_ASYNC_LDS
            __builtin_amdgcn_global_load_async_to_lds_b32(
                (GLOBAL_AS int*)gp, (LDS_AS int*)sp, 0, 0);
#else
            *sp = *gp;
#endif
        }
        wait_async_all();
        __syncthreads();

        #pragma unroll 4
        for (int kk = 0; kk < KCHUNK; kk += 4) {
            const int p = (kk + ko) >> 1;          // K-pair row (kk+ko is even)
            const v2f a  = *reinterpret_cast<const v2f*>(&As[wave * 16 + mrow][kk + ko]);
            const v2f b0 = *reinterpret_cast<const v2f*>(&Ws2[p][(ncol) * 2]);
            const v2f b1 = *reinterpret_cast<const v2f*>(&Ws2[p][(16 + ncol) * 2]);
            const v2f b2 = *reinterpret_cast<const v2f*>(&Ws2[p][(32 + ncol) * 2]);
            acc0 = __builtin_amdgcn_wmma_f32_16x16x4_f32(false, a, false, b0,
                                                         (short)0, acc0, false, false);
            acc1 = __builtin_amdgcn_wmma_f32_16x16x4_f32(false, a, false, b1,
                                                         (short)0, acc1, false, false);
            acc2 = __builtin_amdgcn_wmma_f32_16x16x4_f32(false, a, false, b2,
                                                         (short)0, acc2, false, false);
        }
        __syncthreads();
    }

    // C/D layout: VGPR r, lanes 0-15 -> M=r, lanes 16-31 -> M=r+8, N = lane&15
    const int rowHi = (lane >> 4) * 8;
    #pragma unroll
    for (int r = 0; r < 8; ++r) {
        const int row = m0 + wave * 16 + r + rowHi;
        float* outp = &em[(size_t)row * NTAGS];
        outp[ 0 + ncol] = acc0[r] + bias[ 0 + ncol];
        outp[16 + ncol] = acc1[r] + bias[16 + ncol];
        outp[32 + ncol] = acc2[r] + bias[32 + ncol];
    }
}

// ---------------------------------------------------------------------------
// CRF forward scans. grid (BATCH, 2): y==0 free partition, y==1 constrained.
// 192 threads = 48 tags x 4 quad-lanes; lse over 48 predecessors split 4-way,
// combined with intra-quad shfl_xor (wave32).
// ---------------------------------------------------------------------------
__device__ __forceinline__ float maskVal(int tag, int t)
{
    if (tag == 1)                       // UNCERTAIN: everything except tags 0,1
        return (t == 0 || t == 1) ? 0.0f : 1.0f;
    return (t == tag) ? 1.0f : 0.0f;
}

__global__ __launch_bounds__(192)
void crf_scan_kernel(const float* __restrict__ em,     // [B*S][NTAGS]
                     const int*   __restrict__ tags,   // [B][S]
                     const float* __restrict__ trans,  // [NTAGS][NTAGS]
                     float* __restrict__ partial)      // [2][BATCH]
{
    const int b    = blockIdx.x;
    const int mode = blockIdx.y;
    const int t    = threadIdx.x;
    const int j    = t >> 2;            // output tag 0..47
    const int q    = t & 3;             // quad lane: predecessors i = q + 4u

    __shared__ float lp[NTAGS];
    __shared__ float tr[NTAGS * NTAGS];
    __shared__ float mprev[NTAGS];
    __shared__ float mcur[NTAGS];

    for (int e = t; e < NTAGS * NTAGS; e += 192) tr[e] = trans[e];

    if (mode == 0) {
        if (q == 0) lp[j] = em[((size_t)b * SEQ) * NTAGS + j];
        __syncthreads();
        for (int s = 1; s < SEQ; ++s) {
            const float ej = em[((size_t)b * SEQ + s) * NTAGS + j];
            float sc[12];
            float mx = -3.4e38f;
            #pragma unroll
            for (int u = 0; u < 12; ++u) {
                const int i = q + u * 4;
                sc[u] = lp[i] + tr[i * NTAGS + j];
                mx = fmaxf(mx, sc[u]);
            }
            mx = fmaxf(mx, __shfl_xor(mx, 1));
            mx = fmaxf(mx, __shfl_xor(mx, 2));
            float sum = 0.0f;
            #pragma unroll
            for (int u = 0; u < 12; ++u) sum += __expf(sc[u] - mx);
            sum += __shfl_xor(sum, 1);
            sum += __shfl_xor(sum, 2);
            const float newv = mx + __logf(sum) + ej;
            __syncthreads();
            if (q == 0) lp[j] = newv;
            __syncthreads();
        }
    } else {
        const int tag0 = tags[(size_t)b * SEQ];
        if (q == 0) {
            const float m = maskVal(tag0, j);
            lp[j]    = em[((size_t)b * SEQ) * NTAGS + j] * m + (1.0f - m) * NINF_V;
            mprev[j] = m;
        }
        __syncthreads();
        for (int s = 1; s < SEQ; ++s) {
            const int tg = tags[(size_t)b * SEQ + s];
            const float cm = maskVal(tg, j);
            if (q == 0) mcur[j] = cm;
            const float ej = em[((size_t)b * SEQ + s) * NTAGS + j];
            const float be = ej * cm + (1.0f - cm) * NINF_V;
            float sc[12];
            float mx = -3.4e38f;
            #pragma unroll
            for (int u = 0; u < 12; ++u) {
                const int i  = q + u * 4;
                const float pm  = mprev[i];
                const float blp = lp[i] * pm + (1.0f - pm) * NINF_V;
                const float tm  = pm * cm;
                const float bt  = tr[i * NTAGS + j] * tm + (1.0f - tm) * NINF_V;
                sc[u] = blp + bt + be;
                mx = fmaxf(mx, sc[u]);
            }
            mx = fmaxf(mx, __shfl_xor(mx, 1));
            mx = fmaxf(mx, __shfl_xor(mx, 2));
            float sum = 0.0f;
            #pragma unroll
            for (int u = 0; u < 12; ++u) sum += __expf(sc[u] - mx);
            sum += __shfl_xor(sum, 1);
            sum += __shfl_xor(sum, 2);
            const float newv = mx + __logf(sum);
            __syncthreads();
            if (q == 0) { lp[j] = newv; mprev[j] = mcur[j]; }
            __syncthreads();
        }
    }

    __syncthreads();
    if (t == 0) {
        float mx = lp[0];
        for (int i = 1; i < NTAGS; ++i) mx = fmaxf(mx, lp[i]);
        float sum = 0.0f;
        for (int i = 0; i < NTAGS; ++i) sum += __expf(lp[i] - mx);
        partial[mode * BATCH + b] = mx + __logf(sum);
    }
}

// ---------------------------------------------------------------------------
// Final reduction: out = sum_b (denom[b] - numer[b])
// ---------------------------------------------------------------------------
__global__ __launch_bounds__(64)
void crf_reduce_kernel(const float* __restrict__ partial, float* __restrict__ out)
{
    __shared__ float buf[64];
    const int t = threadIdx.x;
    buf[t] = partial[t] - partial[BATCH + t];   // mode0 (free) - mode1 (constrained)
    __syncthreads();
    for (int off = 32; off > 0; off >>= 1) {
        if (t < off) buf[t] += buf[t + off];
        __syncthreads();
    }
    if (t == 0) out[0] = buf[0];
}

// ---------------------------------------------------------------------------
extern "C" void kernel_launch(void* const* d_in, const int* in_sizes, int n_in,
                              void* d_out, int out_size, void* d_ws, size_t ws_size,
                              hipStream_t stream)
{
    const float* x     = (const float*)d_in[0];
    const int*   y     = (const int*)  d_in[1];
    const float* W     = (const float*)d_in[2];
    const float* bias  = (const float*)d_in[3];
    const float* trans = (const float*)d_in[4];

    float* em      = (float*)d_ws;                          // [B*S][NTAGS]
    float* partial = em + (size_t)BATCH * SEQ * NTAGS;      // [2][BATCH]

    emissions_wmma_kernel<<<dim3((BATCH * SEQ) / ROWS_PER_BLOCK), dim3(128), 0, stream>>>(
        x, W, bias, em);
    crf_scan_kernel<<<dim3(BATCH, 2), dim3(192), 0, stream>>>(em, y, trans, partial);
    crf_reduce_kernel<<<1, 64, 0, stream>>>(partial, (float*)d_out);
}